// GaussianNBNN_21345987461759
// MI455X (gfx1250) — compile-verified
//
#include <hip/hip_runtime.h>
#include <hip/hip_bf16.h>
#include <math.h>

// GaussianNB log-posterior:  out(B,C) = const_c + x^2 @ W1^T + x @ W2^T
//   W1 = -0.5 * exp(-log_var),  W2 = mu * exp(-log_var)
//   const_c = log_pi - 0.5*(F*log2pi + sum log_var + sum mu^2*exp(-log_var))
// B=2048, C=256, F=1024, all fp32. Uses V_WMMA_F32_16X16X4_F32 (full fp32).

typedef __attribute__((ext_vector_type(2))) float v2f;
typedef __attribute__((ext_vector_type(8))) float v8f;

#define GNB_B 2048
#define GNB_C 256
#define GNB_F 1024
#define GNB_LOG2PI 1.837877066409345f

// ---------------------------------------------------------------------------
// Prep: per-class weight vectors and constants. One block per class.
// ---------------------------------------------------------------------------
__global__ __launch_bounds__(256) void gnb_prep_kernel(
    const float* __restrict__ mu,
    const float* __restrict__ log_var,
    const float* __restrict__ log_pi,
    float* __restrict__ w1,        // (C,F)  -0.5*inv_var
    float* __restrict__ w2,        // (C,F)  mu*inv_var
    float* __restrict__ constc)    // (C,)
{
    const int c = blockIdx.x;
    const int t = threadIdx.x;
    const size_t rowoff = (size_t)c * GNB_F;

    float s = 0.0f;   // sum_f (log_var + mu^2 * inv_var)
    for (int f = t; f < GNB_F; f += 256) {
        float lv  = log_var[rowoff + f];
        float m   = mu[rowoff + f];
        float inv = expf(-lv);
        w1[rowoff + f] = -0.5f * inv;
        w2[rowoff + f] = m * inv;
        s += lv + m * m * inv;
    }

    __shared__ float red[256];
    red[t] = s;
    __syncthreads();
    for (int off = 128; off > 0; off >>= 1) {
        if (t < off) red[t] += red[t + off];
        __syncthreads();
    }
    if (t == 0)
        constc[c] = log_pi[c] - 0.5f * ((float)GNB_F * GNB_LOG2PI + red[0]);
}

// ---------------------------------------------------------------------------
// Fused dual-GEMM via V_WMMA_F32_16X16X4_F32.
// Each wave: 64(M) x 16(N) strip = 4 accumulators; 2 WMMAs per tile per K4.
// A (16x4 f32): lane m=lane%16, VGPR pair = K {2*(lane/16), +1}
// B (4x16 f32): lane n=lane%16, VGPR pair = K {2*(lane/16), +1}
// C (16x16 f32): VGPR j -> row j + 8*(lane/16), col = lane%16
// ---------------------------------------------------------------------------
__global__ __launch_bounds__(256) void gnb_wmma_kernel(
    const float* __restrict__ x,
    const float* __restrict__ w1,
    const float* __restrict__ w2,
    const float* __restrict__ constc,
    float* __restrict__ out)
{
    const int lane = threadIdx.x & 31;
    const int wave = (blockIdx.x * 256 + threadIdx.x) >> 5;

    const int nTilesN = GNB_C / 16;            // 16
    const int m64 = wave / nTilesN;            // 0..31  -> 64-row strip
    const int n16 = wave % nTilesN;            // 0..15  -> 16-col tile
    const int b0 = m64 * 64;
    const int c0 = n16 * 16;

    const int r  = lane & 15;                  // row for A loads / col of C tile
    const int kk = lane >> 4;                  // K sub-pair selector

    const float* w1p = w1 + (size_t)(c0 + r) * GNB_F + 2 * kk;
    const float* w2p = w2 + (size_t)(c0 + r) * GNB_F + 2 * kk;
    const float* xp  = x  + (size_t)(b0 + r) * GNB_F + 2 * kk;

    v8f acc0 = {}, acc1 = {}, acc2 = {}, acc3 = {};

#pragma unroll 4
    for (int k = 0; k < GNB_F; k += 4) {
        v2f bw1 = *(const v2f*)(w1p + k);
        v2f bw2 = *(const v2f*)(w2p + k);
        v2f a0  = *(const v2f*)(xp + k);
        v2f a1  = *(const v2f*)(xp + (size_t)16 * GNB_F + k);
        v2f a2  = *(const v2f*)(xp + (size_t)32 * GNB_F + k);
        v2f a3  = *(const v2f*)(xp + (size_t)48 * GNB_F + k);
        v2f q0 = a0 * a0;
        v2f q1 = a1 * a1;
        v2f q2 = a2 * a2;
        v2f q3 = a3 * a3;
        // quad term: x^2 @ (-0.5*inv_var)
        acc0 = __builtin_amdgcn_wmma_f32_16x16x4_f32(false, q0, false, bw1, (short)0, acc0, false, false);
        acc1 = __builtin_amdgcn_wmma_f32_16x16x4_f32(false, q1, false, bw1, (short)0, acc1, false, false);
        acc2 = __builtin_amdgcn_wmma_f32_16x16x4_f32(false, q2, false, bw1, (short)0, acc2, false, false);
        acc3 = __builtin_amdgcn_wmma_f32_16x16x4_f32(false, q3, false, bw1, (short)0, acc3, false, false);
        // cross term: x @ (mu*inv_var)
        acc0 = __builtin_amdgcn_wmma_f32_16x16x4_f32(false, a0, false, bw2, (short)0, acc0, false, false);
        acc1 = __builtin_amdgcn_wmma_f32_16x16x4_f32(false, a1, false, bw2, (short)0, acc1, false, false);
        acc2 = __builtin_amdgcn_wmma_f32_16x16x4_f32(false, a2, false, bw2, (short)0, acc2, false, false);
        acc3 = __builtin_amdgcn_wmma_f32_16x16x4_f32(false, a3, false, bw2, (short)0, acc3, false, false);
    }

    // Epilogue: every C-element this lane holds lives in column (c0 + r),
    // so the bias is one scalar per lane.
    const float bias = constc[c0 + r];

#pragma unroll
    for (int j = 0; j < 8; ++j) {
        const int row = j + 8 * kk;
        out[(size_t)(b0 +      row) * GNB_C + c0 + r] = acc0[j] + bias;
        out[(size_t)(b0 + 16 + row) * GNB_C + c0 + r] = acc1[j] + bias;
        out[(size_t)(b0 + 32 + row) * GNB_C + c0 + r] = acc2[j] + bias;
        out[(size_t)(b0 + 48 + row) * GNB_C + c0 + r] = acc3[j] + bias;
    }
}

// ---------------------------------------------------------------------------
extern "C" void kernel_launch(void* const* d_in, const int* in_sizes, int n_in,
                              void* d_out, int out_size, void* d_ws, size_t ws_size,
                              hipStream_t stream)
{
    const float* x       = (const float*)d_in[0];   // (B,F)
    const float* mu      = (const float*)d_in[1];   // (C,F)
    const float* log_var = (const float*)d_in[2];   // (C,F)
    const float* log_pi  = (const float*)d_in[3];   // (C,)
    float* out = (float*)d_out;                      // (B,C)

    float* w1 = (float*)d_ws;                        // C*F floats
    float* w2 = w1 + (size_t)GNB_C * GNB_F;          // C*F floats
    float* cc = w2 + (size_t)GNB_C * GNB_F;          // C floats

    gnb_prep_kernel<<<GNB_C, 256, 0, stream>>>(mu, log_var, log_pi, w1, w2, cc);

    const int waves  = (GNB_B / 64) * (GNB_C / 16);  // 512
    const int blocks = waves / 8;                    // 256 thr = 8 waves/block
    gnb_wmma_kernel<<<blocks, 256, 0, stream>>>(x, w1, w2, cc, out);
}